// GCNSpatialBlock_57724360458589
// MI455X (gfx1250) — compile-verified
//
#include <hip/hip_runtime.h>

// ---------------------------------------------------------------------------
// SGN-style GCN spatial block for MI455X (gfx1250, wave32, WMMA + TDM).
// All matmuls use V_WMMA_F32_16X16X4_F32 (fp32-faithful; problem is HBM-bound
// at 23.3 TB/s so low precision buys nothing).  Channel-last intermediate
// layout [n,t,v,c] makes every conv a single coalesced GEMM.
// GEMM blocks own 64 rows x all COUT: activations staged once in LDS via the
// Tensor Data Mover (hardware row padding = LDS bank stagger), BN statistics
// computed from accumulators (no extra U pass), deterministic fixed-order
// reductions (no float atomics).
// ---------------------------------------------------------------------------

typedef __attribute__((ext_vector_type(2))) float v2f;
typedef __attribute__((ext_vector_type(8))) float v8f;

__device__ __forceinline__ v8f wmma_f32(v2f a, v2f b, v8f c) {
  // D = A(16x4) * B(4x16) + C(16x16), fp32
  return __builtin_amdgcn_wmma_f32_16x16x4_f32(
      false, a, false, b, (short)0, c, false, false);
}

#define NB   32
#define CIN  128
#define VJ   25
#define TT   256
#define NT   (NB * TT)            // 8192 (n,t) slices
#define NPOS (NB * TT * VJ)       // 204800 positions
#define NMB  (NPOS / 64)          // 3200 GEMM row-blocks
#define X3SZ ((size_t)NB * 256 * VJ * TT)   // 52,428,800 floats

#if __has_builtin(__builtin_amdgcn_tensor_load_to_lds)
#define USE_TDM 1
typedef __attribute__((ext_vector_type(4))) unsigned int u32x4;
typedef __attribute__((ext_vector_type(8))) int i32x8;
typedef __attribute__((ext_vector_type(4))) int i32x4;

// TDM 2D tile load: 64 rows x kdw dwords, row stride kdw dwords in memory,
// LDS rows padded by 4 dwords (pad_interval = kdw, pad_amount = 4) so the
// LDS image has pitch kdw+4 == bank-staggered layout the WMMA loop expects.
// icode = log2(kdw) - 1  (pad_interval encoding: 6 -> 128 dw, 7 -> 256 dw).
__device__ __forceinline__ void tdm_load_2d(unsigned lds_off, const float* gsrc,
                                            unsigned kdw, unsigned icode) {
  unsigned long long ga = (unsigned long long)(size_t)gsrc;
  u32x4 g0;
  g0[0] = 1u;                                                // count=1, user D#
  g0[1] = lds_off;                                           // lds_addr (bytes)
  g0[2] = (unsigned)ga;                                      // global_addr[31:0]
  g0[3] = (unsigned)((ga >> 32) & 0x01ffffffu) | (2u << 30); // [56:32] | type=2
  i32x8 g1;
  g1[0] = (int)((2u << 16)        // data_size = 4 bytes
              | (1u << 20)        // pad_enable
              | (icode << 22)     // pad_interval = kdw dwords
              | (3u << 25));      // pad_amount  = 4 dwords
  g1[1] = (int)((kdw & 0xffffu) << 16);                      // tensor_dim0 lo16
  g1[2] = (int)(((kdw >> 16) & 0xffffu) | (64u << 16));      // dim0 hi | dim1 lo (64 rows)
  g1[3] = (int)((kdw & 0xffffu) << 16);                      // dim1 hi=0 | tile_dim0 = kdw
  g1[4] = (int)64u;                                          // tile_dim1=64, tile_dim2=0
  g1[5] = (int)kdw;                                          // tensor_dim0_stride lo32
  g1[6] = 0;
  g1[7] = 0;                                                 // dim1_stride unused (2D)
  i32x4 z4 = {0, 0, 0, 0};
  i32x8 z8 = {0, 0, 0, 0, 0, 0, 0, 0};
  // 6-arg form (clang-23 / therock-10.0): groups 0..3 + extra group + cpol
  __builtin_amdgcn_tensor_load_to_lds(g0, g1, z4, z4, z8, 0);
}
#else
#define USE_TDM 0
#endif

// ---------------- weight transpose: WT[c*pitch + off + o] = W[o*C + c] ------
__global__ __launch_bounds__(256) void wtrans_kernel(
    const float* __restrict__ W, float* __restrict__ WT,
    int O, int C, int pitch, int coloff) {
  int i = blockIdx.x * 256 + threadIdx.x;
  if (i < O * C) {
    int o = i / C, c = i % C;
    WT[(size_t)c * pitch + coloff + o] = W[i];
  }
}

// ---------------- x[n,c,v,t] -> xt[n,t,v,c] via LDS tile --------------------
__global__ __launch_bounds__(256) void transpose_x_kernel(
    const float* __restrict__ X, float* __restrict__ XT) {
  extern __shared__ float lds[];               // 128 x 65
  int b = blockIdx.x;                          // N*V*4 = 3200
  int n = b / 100, rem = b % 100, v = rem / 4, t0 = (rem % 4) * 64;
  int tid = threadIdx.x;
  int tl = tid & 63, cg = tid >> 6;
#pragma unroll 4
  for (int ci = 0; ci < 32; ++ci) {
    int c = cg * 32 + ci;
    lds[c * 65 + tl] = X[(((size_t)n * CIN + c) * VJ + v) * TT + t0 + tl];
  }
  __syncthreads();
  int cl = tid & 127, tg = tid >> 7;
#pragma unroll 4
  for (int ti = 0; ti < 32; ++ti) {
    int t = tg * 32 + ti;
    XT[(((size_t)n * TT + t0 + t) * VJ + v) * CIN + cl] = lds[cl * 65 + t];
  }
}

// ---------------- fused projection + Gram + softmax per (n,t) ---------------
// AB[32x512] = Xs[32x128] * WTG[128x512]; S = A*B^T (K=256); softmax rows.
__global__ __launch_bounds__(256) void proj_gmat_kernel(
    const float* __restrict__ XT, const float* __restrict__ WTG,
    float* __restrict__ Gout) {
  extern __shared__ float sm[];
  float* Xs = sm;                    // 32*132
  float* AB = sm + 32 * 132;         // 32*516
  float* S  = AB + 32 * 516;         // 32*33
  const int nt = blockIdx.x;
  const size_t p0 = (size_t)nt * VJ;
  const int tid = threadIdx.x, lane = tid & 31, wave = tid >> 5;
  const int hl = lane >> 4, lm = lane & 15;

  for (int i = tid; i < 32 * 132; i += 256) {
    int v = i / 132, c = i % 132;
    Xs[i] = (v < VJ && c < CIN) ? XT[(p0 + v) * CIN + c] : 0.f;
  }
  __syncthreads();

  // phase 2: AB = Xs * WTG (M=32, N=512, K=128); 64 tiles over 8 waves
  for (int tt2 = wave; tt2 < 64; tt2 += 8) {
    int mt = tt2 & 1, ot = tt2 >> 1;
    v8f acc = {};
    for (int k0 = 0; k0 < 128; k0 += 4) {
      int kk = k0 + hl * 2;
      v2f a, b;
      a.x = Xs[(mt * 16 + lm) * 132 + kk];
      a.y = Xs[(mt * 16 + lm) * 132 + kk + 1];
      b.x = WTG[(size_t)kk * 512 + ot * 16 + lm];
      b.y = WTG[(size_t)(kk + 1) * 512 + ot * 16 + lm];
      acc = wmma_f32(a, b, acc);
    }
#pragma unroll
    for (int r = 0; r < 8; ++r)
      AB[(mt * 16 + hl * 8 + r) * 516 + ot * 16 + lm] = acc[r];
  }
  __syncthreads();

  // phase 3: S[v,w] = sum_c A[v,c]*B[w,c]; A=AB[:, :256], B=AB[:, 256:]
  if (wave < 4) {
    int mt = wave & 1, wt2 = wave >> 1;
    v8f acc = {};
    for (int k0 = 0; k0 < 256; k0 += 4) {
      int kk = k0 + hl * 2;
      v2f a, b;
      a.x = AB[(mt * 16 + lm) * 516 + kk];
      a.y = AB[(mt * 16 + lm) * 516 + kk + 1];
      b.x = AB[(wt2 * 16 + lm) * 516 + 256 + kk];
      b.y = AB[(wt2 * 16 + lm) * 516 + 256 + kk + 1];
      acc = wmma_f32(a, b, acc);
    }
#pragma unroll
    for (int r = 0; r < 8; ++r)
      S[(mt * 16 + hl * 8 + r) * 33 + wt2 * 16 + lm] = acc[r];
  }
  __syncthreads();

  if (tid < VJ) {
    int v = tid;
    float mx = -1e30f;
#pragma unroll
    for (int w = 0; w < VJ; ++w) mx = fmaxf(mx, S[v * 33 + w]);
    float e[VJ], sum = 0.f;
#pragma unroll
    for (int w = 0; w < VJ; ++w) { e[w] = __expf(S[v * 33 + w] - mx); sum += e[w]; }
    float inv = 1.f / sum;
#pragma unroll
    for (int w = 0; w < VJ; ++w)
      Gout[(size_t)nt * 625 + v * 25 + w] = e[w] * inv;
  }
}

// ---------------- graph apply per (n,t): Y[v,c] = sum_w G[v,w] X[w,c] -------
__global__ __launch_bounds__(256) void graph_apply_kernel(
    const float* __restrict__ X, const float* __restrict__ G,
    float* __restrict__ Y, int C) {
  const int nt = blockIdx.x;
  const size_t base = (size_t)nt * VJ;
  const size_t gbase = (size_t)nt * 625;
  const int tid = threadIdx.x, lane = tid & 31, wave = tid >> 5;
  const int hl = lane >> 4, lm = lane & 15;
  const int ntiles = 2 * (C >> 4);
  for (int tt2 = wave; tt2 < ntiles; tt2 += 8) {
    const int mt = tt2 & 1, ct = tt2 >> 1;
    const int v = mt * 16 + lm;
    const int c = ct * 16 + lm;
    v8f acc = {};
#pragma unroll
    for (int k0 = 0; k0 < 28; k0 += 4) {
      int w0 = k0 + hl * 2;
      v2f a, b;
      a.x = (v < VJ && w0     < VJ) ? G[gbase + v * 25 + w0]     : 0.f;
      a.y = (v < VJ && w0 + 1 < VJ) ? G[gbase + v * 25 + w0 + 1] : 0.f;
      b.x = (w0     < VJ) ? X[(base + w0)     * C + c] : 0.f;
      b.y = (w0 + 1 < VJ) ? X[(base + w0 + 1) * C + c] : 0.f;
      acc = wmma_f32(a, b, acc);
    }
#pragma unroll
    for (int r = 0; r < 8; ++r) {
      int vr = mt * 16 + hl * 8 + r;
      if (vr < VJ) Y[(base + vr) * C + ct * 16 + lm] = acc[r];
    }
  }
}

// ---------------- big GEMM: U = A1*W1t (+ A2*W2t) ---------------------------
// One block = 64 rows x all COUT columns.  A1/A2 staged once into LDS by the
// Tensor Data Mover (single HBM read of the activation streams, hardware row
// padding to pitch K+4); o-blocks of 128 columns looped inside.  BN partial
// sums (sum, sum^2 per channel) from accumulators, fixed-order -> bit-exact.
__global__ __launch_bounds__(256) void gemm_unit_kernel(
    const float* __restrict__ A1, const float* __restrict__ W1,
    const float* __restrict__ A2, const float* __restrict__ W2,
    float* __restrict__ U, float* __restrict__ part, int K, int COUT) {
  extern __shared__ float lds[];
  const int pitch = K + 4;                  // 16B-aligned rows, bank-staggered
  float* As1 = lds;
  float* As2 = lds + 64 * pitch;
  const int tid = threadIdx.x, lane = tid & 31, wave = tid >> 5;
  const int hl = lane >> 4, lm = lane & 15;
  const size_t m0 = (size_t)blockIdx.x * 64;
  const int K4 = K >> 2;

#if USE_TDM
  if (wave == 0) {
    unsigned icode = (K == 128) ? 6u : 7u;   // pad every K dwords
    tdm_load_2d((unsigned)(size_t)(void*)As1, A1 + m0 * K, (unsigned)K, icode);
    if (A2)
      tdm_load_2d((unsigned)(size_t)(void*)As2, A2 + m0 * K, (unsigned)K, icode);
    __builtin_amdgcn_s_wait_tensorcnt(0);
  }
#else
  for (int i = tid; i < 64 * K4; i += 256) {
    int row = i / K4, c4 = (i - row * K4) * 4;
    *(float4*)(As1 + row * pitch + c4) = *(const float4*)(A1 + (m0 + row) * K + c4);
  }
  if (A2) {
    for (int i = tid; i < 64 * K4; i += 256) {
      int row = i / K4, c4 = (i - row * K4) * 4;
      *(float4*)(As2 + row * pitch + c4) = *(const float4*)(A2 + (m0 + row) * K + c4);
    }
  }
#endif
  __syncthreads();

  const int nsrc = A2 ? 2 : 1;
  for (int ob0 = 0; ob0 < COUT; ob0 += 128) {
    const int o = ob0 + wave * 16 + lm;     // this wave's 16-channel column
    v8f acc0 = {}, acc1 = {}, acc2 = {}, acc3 = {};
    for (int s = 0; s < nsrc; ++s) {
      const float* As = s ? As2 : As1;
      const float* W  = s ? W2 : W1;
      for (int k0 = 0; k0 < K; k0 += 4) {
        int kk = k0 + hl * 2;
        v2f b;
        b.x = W[(size_t)kk * COUT + o];
        b.y = W[(size_t)(kk + 1) * COUT + o];
        v2f a0 = *(const v2f*)(As + (     lm) * pitch + kk);
        v2f a1 = *(const v2f*)(As + (16 + lm) * pitch + kk);
        v2f a2 = *(const v2f*)(As + (32 + lm) * pitch + kk);
        v2f a3 = *(const v2f*)(As + (48 + lm) * pitch + kk);
        acc0 = wmma_f32(a0, b, acc0);
        acc1 = wmma_f32(a1, b, acc1);
        acc2 = wmma_f32(a2, b, acc2);
        acc3 = wmma_f32(a3, b, acc3);
      }
    }
#pragma unroll
    for (int r = 0; r < 8; ++r) {
      U[(m0 +      hl * 8 + r) * COUT + o] = acc0[r];
      U[(m0 + 16 + hl * 8 + r) * COUT + o] = acc1[r];
      U[(m0 + 32 + hl * 8 + r) * COUT + o] = acc2[r];
      U[(m0 + 48 + hl * 8 + r) * COUT + o] = acc3[r];
    }
    if (part) {
      float s1 = 0.f, q1 = 0.f;
#pragma unroll
      for (int r = 0; r < 8; ++r) {
        s1 += acc0[r] + acc1[r] + acc2[r] + acc3[r];
        q1 += acc0[r] * acc0[r] + acc1[r] * acc1[r]
            + acc2[r] * acc2[r] + acc3[r] * acc3[r];
      }
      float s2 = __shfl_down(s1, 16, 32);   // other half: rows +8
      float q2 = __shfl_down(q1, 16, 32);
      if (lane < 16) {
        part[(size_t)blockIdx.x * 512 + o]       = s1 + s2;
        part[(size_t)blockIdx.x * 512 + 256 + o] = q1 + q2;
      }
    }
  }
}

// ---------------- deterministic BN finalize over 3200 partials --------------
__global__ __launch_bounds__(256) void bn_finalize_kernel(
    const float* __restrict__ part, float* __restrict__ stats, int COUT) {
  int c = threadIdx.x;
  if (c >= COUT) return;
  float s = 0.f, q = 0.f;
  for (int b = 0; b < NMB; ++b) {
    s += part[(size_t)b * 512 + c];
    q += part[(size_t)b * 512 + 256 + c];
  }
  const float inv = 1.f / (float)NPOS;
  float m = s * inv;
  float var = q * inv - m * m;
  stats[512 + c] = m;
  stats[768 + c] = __frsqrt_rn(var + 1e-5f);
}

// ---------------- BN apply + relu + residual (channel-last, vec4) -----------
__global__ __launch_bounds__(256) void bn_apply_kernel(
    const float* __restrict__ U, const float* __restrict__ RES,
    const float* __restrict__ stats, const float* __restrict__ gamma,
    const float* __restrict__ beta, float* __restrict__ Xn, int COUT) {
  size_t i = ((size_t)blockIdx.x * 256 + threadIdx.x) * 4;
  int c0 = (int)(i & (size_t)(COUT - 1));
  const float* mean = stats + 512;
  const float* rstd = stats + 768;
  float4 u = *(const float4*)(U + i);
  float4 r = *(const float4*)(RES + i);
  float4 o;
  o.x = fmaxf((u.x - mean[c0    ]) * rstd[c0    ] * gamma[c0    ] + beta[c0    ], 0.f) + r.x;
  o.y = fmaxf((u.y - mean[c0 + 1]) * rstd[c0 + 1] * gamma[c0 + 1] + beta[c0 + 1], 0.f) + r.y;
  o.z = fmaxf((u.z - mean[c0 + 2]) * rstd[c0 + 2] * gamma[c0 + 2] + beta[c0 + 2], 0.f) + r.z;
  o.w = fmaxf((u.w - mean[c0 + 3]) * rstd[c0 + 3] * gamma[c0 + 3] + beta[c0 + 3], 0.f) + r.w;
  *(float4*)(Xn + i) = o;
}

// ---------------- final BN apply + transpose back to [n,c,v,t] --------------
__global__ __launch_bounds__(256) void bn_apply_final_kernel(
    const float* __restrict__ U, const float* __restrict__ RES,
    const float* __restrict__ stats, const float* __restrict__ gamma,
    const float* __restrict__ beta, float* __restrict__ OUT) {
  extern __shared__ float lds[];            // 64 x 257
  int b = blockIdx.x;                       // 3200
  int n = b / 100, rem = b % 100, v = rem / 4, t0 = (rem % 4) * 64;
  int tid = threadIdx.x;
  const float* mean = stats + 512;
  const float* rstd = stats + 768;
  int c = tid;
  float mc = mean[c], rc = rstd[c], gc = gamma[c], bc = beta[c];
  for (int t = 0; t < 64; ++t) {
    size_t p = ((size_t)n * TT + t0 + t) * VJ + v;
    float u = U[p * 256 + c];
    lds[t * 257 + c] = fmaxf((u - mc) * rc * gc + bc, 0.f) + RES[p * 256 + c];
  }
  __syncthreads();
  int tl = tid & 63, cg2 = tid >> 6;
  for (int ci = 0; ci < 64; ++ci) {
    int co = cg2 * 64 + ci;
    OUT[(((size_t)n * 256 + co) * VJ + v) * TT + t0 + tl] = lds[tl * 257 + co];
  }
}

// ---------------------------------------------------------------------------
extern "C" void kernel_launch(void* const* d_in, const int* in_sizes, int n_in,
                              void* d_out, int out_size, void* d_ws, size_t ws_size,
                              hipStream_t stream) {
  const float* x      = (const float*)d_in[0];
  const float* g1w    = (const float*)d_in[1];
  const float* g2w    = (const float*)d_in[2];
  const float* w1_1   = (const float*)d_in[3];
  const float* w2_1   = (const float*)d_in[4];
  const float* gamma1 = (const float*)d_in[5];
  const float* beta1  = (const float*)d_in[6];
  const float* w1_2   = (const float*)d_in[7];
  const float* w2_2   = (const float*)d_in[8];
  const float* gamma2 = (const float*)d_in[9];
  const float* beta2  = (const float*)d_in[10];
  const float* res2w  = (const float*)d_in[11];
  const float* w1_3   = (const float*)d_in[12];
  const float* w2_3   = (const float*)d_in[13];
  const float* gamma3 = (const float*)d_in[14];
  const float* beta3  = (const float*)d_in[15];

  float* out   = (float*)d_out;
  float* g_out = out + X3SZ;
  float* ws    = (float*)d_ws;

  // workspace layout (floats); total ~159.3M floats ~= 637 MB
  float* WTG   = ws;                       // 128 x 512 (g1t | g2t)
  float* WT11  = ws + 65536;               // 128 x 128
  float* WT21  = ws + 81920;
  float* WT12  = ws + 98304;               // 128 x 256
  float* WT22  = ws + 131072;
  float* WTR2  = ws + 163840;
  float* WT13  = ws + 196608;              // 256 x 256
  float* WT23  = ws + 262144;
  float* STATS = ws + 327680;              // 3 x 1024
  float* PART  = ws + 331776;              // 3200 x 512
  float* XT    = ws + 1970176;             // 204800 x 128
  float* X1    = XT + 26214400;            // 204800 x 128
  float* X2    = XT;                       // 204800 x 256 (overlays XT+X1)
  float* Y     = ws + 54398976;            // 204800 x 256 (also R)
  float* U     = ws + 106827776;           // 204800 x 256

  // weight transposes (tiny)
  wtrans_kernel<<<(256*128+255)/256, 256, 0, stream>>>(g1w,  WTG,  256, 128, 512, 0);
  wtrans_kernel<<<(256*128+255)/256, 256, 0, stream>>>(g2w,  WTG,  256, 128, 512, 256);
  wtrans_kernel<<<(128*128+255)/256, 256, 0, stream>>>(w1_1, WT11, 128, 128, 128, 0);
  wtrans_kernel<<<(128*128+255)/256, 256, 0, stream>>>(w2_1, WT21, 128, 128, 128, 0);
  wtrans_kernel<<<(256*128+255)/256, 256, 0, stream>>>(w1_2, WT12, 256, 128, 256, 0);
  wtrans_kernel<<<(256*128+255)/256, 256, 0, stream>>>(w2_2, WT22, 256, 128, 256, 0);
  wtrans_kernel<<<(256*128+255)/256, 256, 0, stream>>>(res2w,WTR2, 256, 128, 256, 0);
  wtrans_kernel<<<(256*256+255)/256, 256, 0, stream>>>(w1_3, WT13, 256, 256, 256, 0);
  wtrans_kernel<<<(256*256+255)/256, 256, 0, stream>>>(w2_3, WT23, 256, 256, 256, 0);

  // x -> channel-last
  transpose_x_kernel<<<3200, 256, 128 * 65 * 4, stream>>>(x, XT);

  // fused projection + Gram + softmax -> g (written straight into d_out)
  proj_gmat_kernel<<<NT, 256, (32*132 + 32*516 + 32*33) * 4, stream>>>(XT, WTG, g_out);

  const int lds1 = 2 * 64 * (128 + 4) * 4;   // dual-source, K=128
  const int ldsR = 1 * 64 * (128 + 4) * 4;   // single-source, K=128
  const int lds3 = 2 * 64 * (256 + 4) * 4;   // dual-source, K=256

  // ---- unit 1: 128 -> 128, residual = x ----
  graph_apply_kernel<<<NT, 256, 0, stream>>>(XT, g_out, Y, 128);
  gemm_unit_kernel<<<NMB, 256, lds1, stream>>>(Y, WT11, XT, WT21, U, PART, 128, 128);
  bn_finalize_kernel<<<1, 256, 0, stream>>>(PART, STATS, 128);
  bn_apply_kernel<<<(int)(NPOS * 128 / 4 / 256), 256, 0, stream>>>(
      U, XT, STATS, gamma1, beta1, X1, 128);

  // ---- unit 2: 128 -> 256, residual = res2_w * x1 ----
  graph_apply_kernel<<<NT, 256, 0, stream>>>(X1, g_out, Y, 128);
  gemm_unit_kernel<<<NMB, 256, lds1, stream>>>(Y, WT12, X1, WT22, U, PART, 128, 256);
  gemm_unit_kernel<<<NMB, 256, ldsR, stream>>>(X1, WTR2, nullptr, nullptr, Y, nullptr, 128, 256); // R -> Y
  bn_finalize_kernel<<<1, 256, 0, stream>>>(PART, STATS + 1024, 256);
  bn_apply_kernel<<<(int)(NPOS * 256 / 4 / 256), 256, 0, stream>>>(
      U, Y, STATS + 1024, gamma2, beta2, X2, 256);

  // ---- unit 3: 256 -> 256, residual = x2; output transposed to [n,c,v,t] ----
  graph_apply_kernel<<<NT, 256, 0, stream>>>(X2, g_out, Y, 256);
  gemm_unit_kernel<<<NMB, 256, lds3, stream>>>(Y, WT13, X2, WT23, U, PART, 256, 256);
  bn_finalize_kernel<<<1, 256, 0, stream>>>(PART, STATS + 2048, 256);
  bn_apply_final_kernel<<<3200, 256, 64 * 257 * 4, stream>>>(
      U, X2, STATS + 2048, gamma3, beta3, out);
}